// ParallelCausalDepthwiseConv1d_481036337261
// MI455X (gfx1250) — compile-verified
//
#include <hip/hip_runtime.h>
#include <hip/hip_bf16.h>

// Causal depthwise conv1d, B=2, C=2048, L=8192, K=4, fp32.
// Bandwidth-bound (~1 FLOP/byte): optimize the data path, not the math.
// CDNA5 path: global_load_async_to_lds_b128 staging + s_wait_asynccnt,
// LDS b128 reads, non-temporal b128 stores.

typedef float v4f __attribute__((ext_vector_type(4)));

constexpr int Bc    = 2;
constexpr int Cc    = 2048;
constexpr int Lc    = 8192;
constexpr int Kc    = 4;

constexpr int BDIM  = 256;                 // 8 waves (wave32)
constexpr int ITEMS = 4;                   // float4 chunks per thread
constexpr int TILE  = BDIM * 4 * ITEMS;    // 4096 floats = 16 KB per block
constexpr int TPR   = Lc / TILE;           // tiles per row = 2

// ---- CDNA5 async global->LDS copy helpers (ASYNCcnt-tracked) ----------------
// ISA 15.18.3: GLOBAL_LOAD_ASYNC_TO_LDS_B128 vdst(lds addr), vaddr(64b), off
// Generic pointer to LDS: high 32 bits = shared aperture, low 32 bits = LDS
// byte address (ISA 10.2 aperture mapping), so truncation yields the LDS addr.

__device__ __forceinline__ unsigned lds_addr32(const void* p) {
  return (unsigned)(unsigned long long)p;
}

__device__ __forceinline__ void async_copy_b128(const float* gsrc, float* ldst) {
  unsigned loff = lds_addr32(ldst);
  asm volatile("global_load_async_to_lds_b128 %0, %1, off"
               :: "v"(loff), "v"(gsrc)
               : "memory");
}

__device__ __forceinline__ void async_copy_b32(const float* gsrc, float* ldst) {
  unsigned loff = lds_addr32(ldst);
  asm volatile("global_load_async_to_lds_b32 %0, %1, off"
               :: "v"(loff), "v"(gsrc)
               : "memory");
}

__device__ __forceinline__ void wait_async0() {
  asm volatile("s_wait_asynccnt 0x0" ::: "memory");
}

// -----------------------------------------------------------------------------

__global__ __launch_bounds__(BDIM)
void ParallelCausalDepthwiseConv1d_481036337261_kernel(
    const float* __restrict__ x,   // [B, C, L]
    const float* __restrict__ w,   // [C, 1, K]
    float* __restrict__ y) {       // [B, C, L]
  // s[0]      : zero pad (never used in math)
  // s[1..3]   : halo x[t0-3 .. t0-1] (zeros at row start)
  // s[4+m]    : x[t0+m], m in [0, TILE)
  __shared__ float s[TILE + 8];

  const int tid = threadIdx.x;
  const int blk = blockIdx.x;
  const int row = blk / TPR;                 // flattened (b, c) row, 0..B*C-1
  const int t0  = (blk % TPR) * TILE;        // tile start within the row
  const int c   = row % Cc;

  const float* __restrict__ xrow = x + (size_t)row * Lc;
  float* __restrict__       yrow = y + (size_t)row * Lc;

  // Stage the main tile: 4 async b128 packets per thread, lane-contiguous
  // 4 KB sweeps per iteration for fully coalesced HBM requests.
#pragma unroll
  for (int u = 0; u < ITEMS; ++u) {
    const int i0 = u * (BDIM * 4) + tid * 4;
    async_copy_b128(xrow + t0 + i0, &s[4 + i0]);
  }

  // Causal halo (3 elements). Uniform branch per block.
  if (t0 == 0) {
    if (tid < 4) s[tid] = 0.0f;              // left edge of the row: zero pad
  } else {
    if (tid < 3) async_copy_b32(xrow + t0 - 3 + tid, &s[1 + tid]);
    if (tid == 3) s[0] = 0.0f;
  }

  wait_async0();        // drain this wave's ASYNCcnt (LDS writes visible)
  __syncthreads();      // make all waves' staged data visible block-wide

  // Per-channel taps: one 16B load, uniform within the block (L2/L0 resident).
  const v4f wv = *(const v4f*)(w + (size_t)c * Kc);
  const float w0 = wv.x, w1 = wv.y, w2 = wv.z, w3 = wv.w;

  // y[t] = w0*x[t-3] + w1*x[t-2] + w2*x[t-1] + w3*x[t]
  // With s[4+m] = x[t0+m]:  x[t0+i-3] = s[i+1]  ->  output i reads s[i+1..i+4].
#pragma unroll
  for (int u = 0; u < ITEMS; ++u) {
    const int i0 = u * (BDIM * 4) + tid * 4;
    const v4f lo = *(const v4f*)&s[i0];      // s[i0 .. i0+3]   (16B aligned)
    const v4f hi = *(const v4f*)&s[i0 + 4];  // s[i0+4 .. i0+7] (16B aligned)
    v4f r;
    r.x = fmaf(w0, lo.y, fmaf(w1, lo.z, fmaf(w2, lo.w, w3 * hi.x)));
    r.y = fmaf(w0, lo.z, fmaf(w1, lo.w, fmaf(w2, hi.x, w3 * hi.y)));
    r.z = fmaf(w0, lo.w, fmaf(w1, hi.x, fmaf(w2, hi.y, w3 * hi.z)));
    r.w = fmaf(w0, hi.x, fmaf(w1, hi.y, fmaf(w2, hi.z, w3 * hi.w)));
    // Streaming output: non-temporal so y doesn't evict x from near caches.
    __builtin_nontemporal_store(r, (v4f*)(yrow + t0 + i0));
  }
}

extern "C" void kernel_launch(void* const* d_in, const int* in_sizes, int n_in,
                              void* d_out, int out_size, void* d_ws, size_t ws_size,
                              hipStream_t stream) {
  (void)in_sizes; (void)n_in; (void)out_size; (void)d_ws; (void)ws_size;
  const float* x = (const float*)d_in[0];   // [B, C, L] fp32
  const float* w = (const float*)d_in[1];   // [C, 1, K] fp32
  float*       y = (float*)d_out;           // [B, C, L] fp32

  const int grid = Bc * Cc * TPR;           // 8192 blocks, one 16KB tile each
  ParallelCausalDepthwiseConv1d_481036337261_kernel<<<grid, BDIM, 0, stream>>>(x, w, y);
}